// CRF_31001073942618
// MI455X (gfx1250) — compile-verified
//
#include <hip/hip_runtime.h>

#define NS    64
#define NOBS  128
#define TLEN  262144
#define NCHK  (TLEN / 64)   // 4096 checkpoints / segments

// ---------------- CDNA5 async-to-LDS helpers (gfx1250) ----------------
// Per-lane async copy global->LDS, tracked by ASYNCcnt (CDNA5 ISA).
__device__ __forceinline__ void async_load_b32(const void* gaddr, void* ldsaddr) {
    // Generic pointer to __shared__ has the LDS byte offset in its low 32 bits.
    unsigned l = (unsigned)(unsigned long long)ldsaddr;
    asm volatile("global_load_async_to_lds_b32 %0, %1, off"
                 :: "v"(l), "v"(gaddr) : "memory");
}
__device__ __forceinline__ void wait_async0() {
    asm volatile("s_wait_asynccnt 0x0" ::: "memory");
}
__device__ __forceinline__ void wait_async1() {
    asm volatile("s_wait_asynccnt 0x1" ::: "memory");
}

// ---------------- Kernel 1: forward Viterbi scan ----------------
// 1 workgroup, 64 threads (2 wave32). Lane tid owns output state i=tid.
// Per step, LDS dsum[j] = delta[j] + ll_t[j] already holds the broadcast vector;
// each lane does a 64-wide max-plus reduction over its register-resident
// transition row. One __syncthreads per step.
__global__ __launch_bounds__(64) void viterbi_fwd(
    const int*   __restrict__ rolls,
    const float* __restrict__ trans,   // [64, 65] row-major
    const float* __restrict__ llg,     // [128, 64] row-major
    float*       __restrict__ out,     // out[TLEN] = final score
    unsigned*    __restrict__ bpws,    // packed backpointers: word (t>>2)*64+i, byte t&3
    int*         __restrict__ cpws,    // checkpoint ancestor tables: [NCHK+1][64]
    int*         __restrict__ fstate)  // final argmax state
{
    __shared__ float             lltab[NOBS * NS];      // 32 KB, whole table in LDS
    __shared__ __align__(16) float dsumbuf[2][NS];
    __shared__ int               ancbuf[2][NS];
    __shared__ int               rollbuf[4][NS];        // 4-deep chunk ring, 64 rolls each
    __shared__ float             dfin[NS];

    const int tid = threadIdx.x;

    // Transition row for this state, resident in VGPRs.
    float trow[NS];
#pragma unroll
    for (int j = 0; j < NS; ++j) trow[j] = trans[tid * (NS + 1) + j];
    const float tinit = trans[tid * (NS + 1) + NS];

    // Kick off async prefetch of roll chunks 0..2 (prefetch distance 2).
    for (int c = 0; c < 3; ++c)
        async_load_b32(&rolls[c * 64 + tid], &rollbuf[c][tid]);

    // Stage the full loglik table into LDS (coalesced).
    for (int k = tid; k < NOBS * NS; k += NS) lltab[k] = llg[k];

    wait_async0();
    __syncthreads();

    // t = 0 init:  delta0 = trans[:,64] + ll[roll0];  dsum for step 1.
    const int r0 = rollbuf[0][0];
    float dcur = tinit + lltab[r0 * NS + tid];
    const int r1 = rollbuf[0][1];
    dsumbuf[0][tid] = dcur + lltab[r1 * NS + tid];
    ancbuf[0][tid]  = tid;
    __syncthreads();

    unsigned pack = 0;
    for (int t = 1; t < TLEN; ++t) {
        if ((t & 63) == 0) {
            // Entering block n = t>>6: issue chunk n+2, ensure chunk n+1 landed.
            const int chunk = (t >> 6) + 2;
            int g = chunk * 64 + tid;
            if (g >= TLEN) g = TLEN - 1;               // clamp (tail garbage unused)
            async_load_b32(&rolls[g], &rollbuf[chunk & 3][tid]);
            wait_async1();                              // oldest (chunk n+1) done
        }

        const int pp = (t - 1) & 1, cp = t & 1;

        // Broadcast-read dsum into registers (16x ds_load_b128).
        float dv[NS];
#pragma unroll
        for (int k = 0; k < NS / 4; ++k)
            ((float4*)dv)[k] = ((const float4*)dsumbuf[pp])[k];

        // Max-plus reduction with first-max-wins argmax (contiguous-range tree:
        // left operand always covers smaller original indices, ties keep left).
        float sv[NS]; int si[NS];
#pragma unroll
        for (int j = 0; j < NS; ++j) { sv[j] = trow[j] + dv[j]; si[j] = j; }
#pragma unroll
        for (int off = 1; off < NS; off <<= 1) {
#pragma unroll
            for (int j = 0; j < NS; j += 2 * off) {
                if (!(sv[j] >= sv[j + off])) { sv[j] = sv[j + off]; si[j] = si[j + off]; }
            }
        }
        const float best = sv[0];
        const int   bi   = si[0];

        // Backpointer: pack 4 steps per u32, coalesced store every 4th step.
        pack |= (unsigned)bi << (8 * (t & 3));
        if ((t & 3) == 3) { bpws[(t >> 2) * NS + tid] = pack; pack = 0; }

        // Ancestor propagation + checkpoint every 64 steps.
        int av = ancbuf[pp][bi];
        if ((t & 63) == 0) { cpws[(t >> 6) * NS + tid] = av; av = tid; }
        ancbuf[cp][tid] = av;

        // Pre-add next step's emission: dsum_next[j] = best + ll_{t+1}[j].
        int r = t + 1; if (r > TLEN - 1) r = TLEN - 1;
        const int roll = rollbuf[(r >> 6) & 3][r & 63];
        dsumbuf[cp][tid] = best + lltab[roll * NS + tid];

        dcur = best;
        __syncthreads();
    }

    // Final ancestor table: maps state@T-1 -> state@(T-64... last checkpoint 262080).
    cpws[NCHK * NS + tid] = ancbuf[(TLEN - 1) & 1][tid];
    dfin[tid] = dcur;
    __syncthreads();

    if (tid == 0) {
        float fs = dfin[0]; int fi = 0;
        for (int j = 1; j < NS; ++j) if (dfin[j] > fs) { fs = dfin[j]; fi = j; }
        fstate[0]  = fi;
        out[TLEN]  = fs;      // final_score
    }
}

// ---------------- Kernel 2: walk the checkpoint chain (4096 dependent loads) ----
__global__ void viterbi_chk(const int* __restrict__ cpws,
                            const int* __restrict__ fstate,
                            int* __restrict__ Sarr)   // Sarr[c] = state at time 64*c
{
    if (threadIdx.x == 0 && blockIdx.x == 0) {
        int s = fstate[0];
        s = cpws[NCHK * NS + s];          // state at time 64*(NCHK-1) = 262080
        Sarr[NCHK - 1] = s;
        for (int c = NCHK - 1; c >= 1; --c) {
            s = cpws[c * NS + s];         // state at time 64*(c-1)
            Sarr[c - 1] = s;
        }
    }
}

// ---------------- Kernel 3: fill all 4096 segments in parallel ----------------
__global__ __launch_bounds__(128) void viterbi_fill(
    const unsigned* __restrict__ bpws,
    const int*      __restrict__ Sarr,
    const int*      __restrict__ fstate,
    float*          __restrict__ out)     // reverse-time path: out[k] = state@(T-1-k)
{
    const int c = blockIdx.x * blockDim.x + threadIdx.x;
    if (c >= NCHK) return;

    int s, thi;
    if (c == NCHK - 1) {
        s = fstate[0];
        out[0] = (float)s;                // time T-1
        thi = TLEN - 1;
    } else {
        s = Sarr[c + 1];                  // state at time 64*(c+1)
        thi = (c + 1) * 64;
    }
    for (int t = thi; t >= c * 64 + 1; --t) {
        const unsigned w = bpws[(t >> 2) * NS + s];
        s = (int)((w >> (8 * (t & 3))) & 0xffu);
        out[TLEN - 1 - (t - 1)] = (float)s;   // state at time t-1
    }
}

// ---------------- Launch ----------------
extern "C" void kernel_launch(void* const* d_in, const int* in_sizes, int n_in,
                              void* d_out, int out_size, void* d_ws, size_t ws_size,
                              hipStream_t stream) {
    const int*   rolls = (const int*)d_in[0];
    const float* trans = (const float*)d_in[1];   // [64,65]
    const float* llg   = (const float*)d_in[2];   // [128,64]
    float* out = (float*)d_out;                   // [TLEN path + 1 score]

    // Workspace layout:
    //   [0, 16MB)        packed backpointers: (TLEN/4)*64 u32 = 16 MiB exactly
    //   +16MB            checkpoint tables:   (NCHK+1)*64 int  (~1 MiB)
    //   after            Sarr: NCHK int; then fstate: 1 int
    unsigned* bpws = (unsigned*)d_ws;
    int* cpws   = (int*)((char*)d_ws + (size_t)16 * 1024 * 1024);
    int* Sarr   = (int*)((char*)cpws + (size_t)(NCHK + 1) * NS * sizeof(int));
    int* fstate = Sarr + NCHK;

    viterbi_fwd <<<1, 64, 0, stream>>>(rolls, trans, llg, out, bpws, cpws, fstate);
    viterbi_chk <<<1, 1, 0, stream>>>(cpws, fstate, Sarr);
    viterbi_fill<<<NCHK / 128, 128, 0, stream>>>(bpws, Sarr, fstate, out);
}